// MultiHeadLiftLayer_31009663877641
// MI455X (gfx1250) — compile-verified
//
#include <hip/hip_runtime.h>
#include <hip/hip_bf16.h>

// ---------------------------------------------------------------------------
// MultiHeadLiftLayer for MI455X (gfx1250, wave32)
//
//   Kernel 1: node projection  p[n] = [ x0[n]·att_k[:128] (k=0..2) | pad |
//                                       x0[n]·att_k[128:] (k=0..2) | pad ]
//             done with V_WMMA_F32_16X16X4_F32 (exact fp32), one wave = one
//             16-node tile, 32 K-steps of 4.
//   Kernel 2: HBM-bound fuse: out[e, 0:3] = relu(s[src[e]] + t[tgt[e]]),
//             out[e, 3:67] = x1[e, :].  One thread per output element.
// ---------------------------------------------------------------------------

typedef float v2f __attribute__((ext_vector_type(2)));
typedef float v8f __attribute__((ext_vector_type(8)));

#define C0_DIM   128
#define K_HEADS  3
#define C1_DIM   64
#define OUT_COLS (K_HEADS + C1_DIM)   // 67
#define P_STRIDE 8                    // [s0,s1,s2,pad, t0,t1,t2,pad]

// ---------------------------------------------------------------------------
// Kernel 1: WMMA fp32 projection. grid: one wave per 16-row node tile.
// ---------------------------------------------------------------------------
__global__ void __launch_bounds__(128)
node_project_wmma(const float* __restrict__ x0,
                  const float* __restrict__ att,     // (3, 256) row-major
                  float* __restrict__ p,             // (N, 8) workspace
                  int nNodes, int nTiles) {
    const int lane = threadIdx.x & 31;
    const int wave = threadIdx.x >> 5;
    const int tile = blockIdx.x * (blockDim.x >> 5) + wave;   // wave-uniform
    if (tile >= nTiles) return;                               // uniform branch

    const int col = lane & 15;   // B / D column index (N dim of WMMA)
    const int hi  = lane >> 4;   // 0: holds K pair {0,1}; 1: K pair {2,3}
    const int row = lane & 15;   // A row within the 16-node tile

    // Branch-free B operand generation from att (keeps EXEC all-ones):
    //   col 0..2  -> att[col][k]          (source half)
    //   col 3..5  -> att[col-3][128 + k]  (target half)
    //   col 6..15 -> 0 (masked)
    int   wbase;
    float wmask;
    if (col < K_HEADS)          { wbase = col * (2 * C0_DIM);                     wmask = 1.0f; }
    else if (col < 2 * K_HEADS) { wbase = (col - K_HEADS) * (2 * C0_DIM) + C0_DIM; wmask = 1.0f; }
    else                        { wbase = 0;                                      wmask = 0.0f; }

    const int nodeRow = tile * 16 + row;
    const int aRow    = (nodeRow < nNodes) ? nodeRow : (nNodes - 1);  // clamp (N%16==0 anyway)
    const float* __restrict__ aPtr = x0 + (size_t)aRow * C0_DIM;

    v8f acc = {0.f, 0.f, 0.f, 0.f, 0.f, 0.f, 0.f, 0.f};

#pragma unroll
    for (int kk = 0; kk < C0_DIM / 4; ++kk) {
        const int k0 = kk * 4 + hi * 2;
        v2f a;                       // A: 16x4 fp32 fragment
        a.x = aPtr[k0];
        a.y = aPtr[k0 + 1];
        v2f b;                       // B: 4x16 fp32 fragment (synthesized W)
        b.x = att[wbase + k0]     * wmask;
        b.y = att[wbase + k0 + 1] * wmask;
        // 8 args: (neg_a, A, neg_b, B, c_mod, C, reuse_a, reuse_b)
        acc = __builtin_amdgcn_wmma_f32_16x16x4_f32(
                  false, a, false, b, (short)0, acc, false, false);
    }

    // D layout: VGPR j -> (lanes 0-15: M=j, N=lane) (lanes 16-31: M=j+8, N=lane-16)
    if (col < 2 * K_HEADS) {
        const int off = (col < K_HEADS) ? col : (col + 1);  // s:0..2  t:4..6
#pragma unroll
        for (int j = 0; j < 8; ++j) {
            const int m    = j + hi * 8;
            const int node = tile * 16 + m;
            if (node < nNodes)
                p[(size_t)node * P_STRIDE + off] = acc[j];
        }
    }
}

// ---------------------------------------------------------------------------
// Kernel 2: HBM-bound gather + copy. One thread per output element.
// total = E*67 = 107.2M < 2^31 -> pure 32-bit indexing, /67 is a magic-mul.
// ---------------------------------------------------------------------------
__global__ void __launch_bounds__(256)
edge_fuse(const float* __restrict__ p,
          const float* __restrict__ x1,
          const int*   __restrict__ esrc,
          const int*   __restrict__ etgt,
          float*       __restrict__ out,
          unsigned total) {
    const unsigned i = blockIdx.x * 256u + threadIdx.x;
    if (i >= total) return;

    const unsigned e = i / (unsigned)OUT_COLS;
    const unsigned c = i - e * (unsigned)OUT_COLS;

    float v;
    if (c < (unsigned)K_HEADS) {
        const unsigned s = (unsigned)esrc[e];
        const unsigned t = (unsigned)etgt[e];
        const float sv = p[(size_t)s * P_STRIDE + c];        // source half
        const float tv = p[(size_t)t * P_STRIDE + 4u + c];   // target half
        v = sv + tv;
        v = (v > 0.0f) ? v : 0.0f;                           // relu
    } else {
        v = x1[(size_t)e * C1_DIM + (c - (unsigned)K_HEADS)];
    }
    out[i] = v;
}

// ---------------------------------------------------------------------------
// Launch
// ---------------------------------------------------------------------------
extern "C" void kernel_launch(void* const* d_in, const int* in_sizes, int n_in,
                              void* d_out, int out_size, void* d_ws, size_t ws_size,
                              hipStream_t stream) {
    const float* x0  = (const float*)d_in[0];   // (N, 128) fp32
    const float* x1  = (const float*)d_in[1];   // (E, 64)  fp32
    const float* att = (const float*)d_in[2];   // (3, 256) fp32
    const int*   ei  = (const int*)d_in[3];     // (2, E)   int32 (JAX x64 off)

    float* out = (float*)d_out;                 // (E, 67) fp32
    float* p   = (float*)d_ws;                  // (N, 8)  fp32 scratch (3.2 MB)

    const int nNodes = in_sizes[0] / C0_DIM;
    const int nEdges = in_sizes[3] / 2;
    const int* esrc  = ei;
    const int* etgt  = ei + nEdges;

    // Kernel 1: projection via fp32 WMMA
    const int nTiles        = (nNodes + 15) / 16;
    const int wavesPerBlock = 4;
    dim3 blockA(32 * wavesPerBlock);
    dim3 gridA((nTiles + wavesPerBlock - 1) / wavesPerBlock);
    node_project_wmma<<<gridA, blockA, 0, stream>>>(x0, att, p, nNodes, nTiles);

    // Kernel 2: bandwidth-bound fuse (depends on p; same stream => ordered)
    const unsigned total = (unsigned)nEdges * (unsigned)OUT_COLS;
    dim3 blockB(256);
    dim3 gridB((total + 255u) / 256u);
    edge_fuse<<<gridB, blockB, 0, stream>>>(p, x1, esrc, etgt, out, total);
}